// GATEncoder_6390911336484
// MI455X (gfx1250) — compile-verified
//
#include <hip/hip_runtime.h>
#include <cmath>
#include <cstdint>

typedef float v2f __attribute__((ext_vector_type(2)));
typedef float v8f __attribute__((ext_vector_type(8)));

#define N_NODES 50000
#define N_EDGES 800000
#define N_GRAPHS 128

// ---------------------------------------------------------------------------
// fill kernel (scratch init every call; harness does not re-poison)
// ---------------------------------------------------------------------------
__global__ void fill_f32(float* __restrict__ p, float val, int n) {
    int i = blockIdx.x * blockDim.x + threadIdx.x;
    if (i < n) p[i] = val;
}

// ---------------------------------------------------------------------------
// GEMM: C[M,Ncol] = A[M,K] @ B[K,Ncol], fp32 via V_WMMA_F32_16X16X4_F32.
// Block = 256 threads = 8 waves; each wave computes one 16x16 tile of C.
// Block covers 128 rows x 16 cols. B column-tile staged in LDS with CDNA5
// async global->LDS copies (ASYNCcnt), bypassing the VGPR file.
// A-fragment (16x4): lane l -> m = l%16, kpair = (l/16)*2, 2 VGPRs = k, k+1.
// B-fragment (4x16): lane l -> n = l%16, kpair = (l/16)*2 (symmetric layout).
// C/D (16x16): vgpr v, lane l -> row = v + 8*(l/16), col = l%16.
// ---------------------------------------------------------------------------
__global__ void gemm_wmma_f32(const float* __restrict__ A,
                              const float* __restrict__ B,
                              float* __restrict__ C,
                              int M, int K, int Ncol) {
    __shared__ float Bs[256 * 16];   // up to K=256 rows of a 16-wide B tile (16 KB)

    const int tid  = threadIdx.x;
    const int lane = tid & 31;
    const int wave = tid >> 5;
    const int col0 = blockIdx.y * 16;
    const int row0 = blockIdx.x * 128 + wave * 16;

    // cooperative ASYNC copy of B tile (K x 16 floats) into LDS.
    // Each lane moves 16B per global_load_async_to_lds_b128 (LDS dest is
    // 16B aligned: i is a multiple of 4 floats). Tracked by ASYNCcnt.
    for (int i = tid * 4; i < K * 16; i += 256 * 4) {
        const float* gp = B + (size_t)(i >> 4) * Ncol + col0 + (i & 15);
        unsigned lds_off = (unsigned)(uintptr_t)(&Bs[i]);
        asm volatile("global_load_async_to_lds_b128 %0, %1, off"
                     :: "v"(lds_off), "v"((unsigned long long)(uintptr_t)gp)
                     : "memory");
    }
    asm volatile("s_wait_asynccnt 0x0" ::: "memory");
    __syncthreads();

    const int half = lane >> 4;          // 0 or 1
    const int ml   = lane & 15;          // m (for A) / n (for B) within tile
    const int kb   = half * 2;           // k sub-offset within the 4-wide step
    int m = row0 + ml;
    if (m >= M) m = M - 1;               // clamp loads; EXEC stays all-1 for WMMA

    const float* arow = A + (size_t)m * K;

    v8f acc = {0.f, 0.f, 0.f, 0.f, 0.f, 0.f, 0.f, 0.f};
    for (int k = 0; k < K; k += 4) {
        v2f a;
        a.x = arow[k + kb];
        a.y = arow[k + kb + 1];
        v2f b;
        b.x = Bs[(k + kb) * 16 + ml];
        b.y = Bs[(k + kb + 1) * 16 + ml];
        acc = __builtin_amdgcn_wmma_f32_16x16x4_f32(
                  false, a, false, b, (short)0, acc, false, false);
    }

#pragma unroll
    for (int v = 0; v < 8; ++v) {
        int r = row0 + v + 8 * half;
        if (r < M) C[(size_t)r * Ncol + col0 + ml] = acc[v];
    }
}

// ---------------------------------------------------------------------------
// per-(node, head) attention logits: a_src = <h[n,h,:], att_src[h,:]>, etc.
// ---------------------------------------------------------------------------
__global__ void attn_scores(const float* __restrict__ h,
                            const float* __restrict__ att_src,
                            const float* __restrict__ att_dst,
                            float* __restrict__ asrc,
                            float* __restrict__ adst,
                            int Nn, int H, int C) {
    int i = blockIdx.x * blockDim.x + threadIdx.x;
    if (i >= Nn * H) return;
    int n  = i / H;
    int hh = i - n * H;
    const float* hp = h + ((size_t)n * H + hh) * C;
    const float* ps = att_src + hh * C;
    const float* pd = att_dst + hh * C;
    float ss = 0.f, sd = 0.f;
    for (int c = 0; c < C; ++c) {
        float v = hp[c];
        ss += v * ps[c];
        sd += v * pd[c];
    }
    asrc[i] = ss;
    adst[i] = sd;
}

// ---------------------------------------------------------------------------
// float atomic max via signed/unsigned int punning (valid for finite floats,
// -inf init). Positive: signed atomicMax. Negative: unsigned atomicMin.
// ---------------------------------------------------------------------------
__device__ __forceinline__ void atomicMaxF32(float* addr, float val) {
    if (val >= 0.0f) {
        atomicMax((int*)addr, __float_as_int(val));
    } else {
        atomicMin((unsigned int*)addr, __float_as_uint(val));
    }
}

__device__ __forceinline__ void edge_decode(const int* __restrict__ ei,
                                            int e, int* s, int* d) {
    if (e < N_EDGES) { *s = ei[e]; *d = ei[N_EDGES + e]; }
    else             { *s = *d = e - N_EDGES; }   // self loop
}

__device__ __forceinline__ float leaky(float v) {
    return v > 0.f ? v : 0.2f * v;
}

// pass 1: segment max of leaky_relu(a_src[src]+a_dst[dst]) over dst
__global__ void edge_max(const int* __restrict__ ei,
                         const float* __restrict__ asrc,
                         const float* __restrict__ adst,
                         float* __restrict__ mx, int H, int ETH) {
    int i = blockIdx.x * blockDim.x + threadIdx.x;
    if (i >= ETH) return;
    int e = i / H, hh = i - e * H;
    int s, d; edge_decode(ei, e, &s, &d);
    float v = leaky(asrc[s * H + hh] + adst[d * H + hh]);
    atomicMaxF32(&mx[d * H + hh], v);
}

// pass 2: segment sum of exp(e - max)
__global__ void edge_expsum(const int* __restrict__ ei,
                            const float* __restrict__ asrc,
                            const float* __restrict__ adst,
                            const float* __restrict__ mx,
                            float* __restrict__ sm, int H, int ETH) {
    int i = blockIdx.x * blockDim.x + threadIdx.x;
    if (i >= ETH) return;
    int e = i / H, hh = i - e * H;
    int s, d; edge_decode(ei, e, &s, &d);
    float v = leaky(asrc[s * H + hh] + adst[d * H + hh]);
    float ee = __expf(v - mx[d * H + hh]);
    atomicAdd(&sm[d * H + hh], ee);
}

// pass 3: out[dst,h,:] += h[src,h,:] * alpha  (alpha = softmax weight)
__global__ void edge_aggregate(const int* __restrict__ ei,
                               const float* __restrict__ asrc,
                               const float* __restrict__ adst,
                               const float* __restrict__ mx,
                               const float* __restrict__ sm,
                               const float* __restrict__ hfeat,
                               float* __restrict__ out,
                               int H, int C, int ETH) {
    int i = blockIdx.x * blockDim.x + threadIdx.x;
    if (i >= ETH) return;
    int e = i / H, hh = i - e * H;
    int s, d; edge_decode(ei, e, &s, &d);
    float v = leaky(asrc[s * H + hh] + adst[d * H + hh]);
    float alpha = __expf(v - mx[d * H + hh]) / (sm[d * H + hh] + 1e-16f);

    const float4* hp = (const float4*)(hfeat + ((size_t)s * H + hh) * C);
    float* op = out + ((size_t)d * H + hh) * C;
    for (int c4 = 0; c4 < C / 4; ++c4) {
        float4 hv = hp[c4];
        int c = c4 * 4;
        atomicAdd(&op[c + 0], hv.x * alpha);
        atomicAdd(&op[c + 1], hv.y * alpha);
        atomicAdd(&op[c + 2], hv.z * alpha);
        atomicAdd(&op[c + 3], hv.w * alpha);
    }
}

// bias + ELU, in place:  p[i] = elu(p[i] + b[i % cols])
__global__ void bias_elu(float* __restrict__ p, const float* __restrict__ b,
                         int n, int cols) {
    int i = blockIdx.x * blockDim.x + threadIdx.x;
    if (i >= n) return;
    float v = p[i] + b[i % cols];
    p[i] = v > 0.f ? v : (__expf(v) - 1.f);
}

// add bias2 and accumulate per-graph sums
__global__ void pool_accum(const float* __restrict__ agg,
                           const float* __restrict__ b2,
                           const int* __restrict__ batch,
                           float* __restrict__ pool, int Nn, int C) {
    int i = blockIdx.x * blockDim.x + threadIdx.x;
    if (i >= Nn * C) return;
    int n = i / C, c = i - n * C;
    atomicAdd(&pool[(size_t)batch[n] * C + c], agg[i] + b2[c]);
}

__global__ void count_nodes(const int* __restrict__ batch,
                            float* __restrict__ cnt, int Nn) {
    int i = blockIdx.x * blockDim.x + threadIdx.x;
    if (i < Nn) atomicAdd(&cnt[batch[i]], 1.0f);
}

__global__ void pool_div(const float* __restrict__ pool,
                         const float* __restrict__ cnt,
                         float* __restrict__ out, int GC, int C) {
    int i = blockIdx.x * blockDim.x + threadIdx.x;
    if (i >= GC) return;
    out[i] = pool[i] / fmaxf(cnt[i / C], 1.0f);
}

// ---------------------------------------------------------------------------
extern "C" void kernel_launch(void* const* d_in, const int* in_sizes, int n_in,
                              void* d_out, int out_size, void* d_ws, size_t ws_size,
                              hipStream_t stream) {
    const float* x        = (const float*)d_in[0];   // [N,128]
    const float* W1       = (const float*)d_in[1];   // [128,256]
    const float* att_src1 = (const float*)d_in[2];   // [4,64]
    const float* att_dst1 = (const float*)d_in[3];   // [4,64]
    const float* b1       = (const float*)d_in[4];   // [256]
    const float* W2       = (const float*)d_in[5];   // [256,128]
    const float* att_src2 = (const float*)d_in[6];   // [1,128]
    const float* att_dst2 = (const float*)d_in[7];   // [1,128]
    const float* b2       = (const float*)d_in[8];   // [128]
    const int*   ei       = (const int*)d_in[9];     // [2,E]
    const int*   batch    = (const int*)d_in[10];    // [N]

    const int N = N_NODES, E = N_EDGES, G = N_GRAPHS;
    const int ET = E + N;                // edges + self loops

    // ---- workspace carve (floats) ----
    float* ws    = (float*)d_ws;
    float* H1    = ws;                   // [N,256]  (later reused: H2 + AGG2)
    float* AGG1  = H1    + (size_t)N * 256;  // [N,256]
    float* ASRC1 = AGG1  + (size_t)N * 256;  // [N,4]
    float* ADST1 = ASRC1 + (size_t)N * 4;
    float* M1    = ADST1 + (size_t)N * 4;
    float* S1    = M1    + (size_t)N * 4;
    float* ASRC2 = S1    + (size_t)N * 4;    // [N]
    float* ADST2 = ASRC2 + N;
    float* M2    = ADST2 + N;
    float* S2    = M2    + N;
    float* PCNT  = S2    + N;                // [G]
    float* POOL  = PCNT  + G;                // [G,128]
    float* H2    = H1;                       // [N,128] reuse
    float* AGG2  = H1 + (size_t)N * 128;     // [N,128] reuse

    const int B = 256;
    const float NEG_INF = -INFINITY;

    // ===== Layer 1 =====
    {   // h1 = x @ W1   (M=N, K=128, Ncol=256)
        dim3 grid((N + 127) / 128, 256 / 16);
        gemm_wmma_f32<<<grid, B, 0, stream>>>(x, W1, H1, N, 128, 256);
    }
    attn_scores<<<(N * 4 + B - 1) / B, B, 0, stream>>>(H1, att_src1, att_dst1,
                                                       ASRC1, ADST1, N, 4, 64);
    fill_f32<<<(N * 4 + B - 1) / B, B, 0, stream>>>(M1, NEG_INF, N * 4);
    fill_f32<<<(N * 4 + B - 1) / B, B, 0, stream>>>(S1, 0.f, N * 4);
    fill_f32<<<(N * 256 + B - 1) / B, B, 0, stream>>>(AGG1, 0.f, N * 256);

    int eth1 = ET * 4;
    edge_max     <<<(eth1 + B - 1) / B, B, 0, stream>>>(ei, ASRC1, ADST1, M1, 4, eth1);
    edge_expsum  <<<(eth1 + B - 1) / B, B, 0, stream>>>(ei, ASRC1, ADST1, M1, S1, 4, eth1);
    edge_aggregate<<<(eth1 + B - 1) / B, B, 0, stream>>>(ei, ASRC1, ADST1, M1, S1,
                                                         H1, AGG1, 4, 64, eth1);
    // elu(agg1 + b1) in place -> input of layer 2
    bias_elu<<<(N * 256 + B - 1) / B, B, 0, stream>>>(AGG1, b1, N * 256, 256);

    // ===== Layer 2 =====
    {   // h2 = elu_out @ W2   (M=N, K=256, Ncol=128)
        dim3 grid((N + 127) / 128, 128 / 16);
        gemm_wmma_f32<<<grid, B, 0, stream>>>(AGG1, W2, H2, N, 256, 128);
    }
    attn_scores<<<(N + B - 1) / B, B, 0, stream>>>(H2, att_src2, att_dst2,
                                                   ASRC2, ADST2, N, 1, 128);
    fill_f32<<<(N + B - 1) / B, B, 0, stream>>>(M2, NEG_INF, N);
    fill_f32<<<(N + B - 1) / B, B, 0, stream>>>(S2, 0.f, N);
    fill_f32<<<(N * 128 + B - 1) / B, B, 0, stream>>>(AGG2, 0.f, N * 128);

    int eth2 = ET;
    edge_max     <<<(eth2 + B - 1) / B, B, 0, stream>>>(ei, ASRC2, ADST2, M2, 1, eth2);
    edge_expsum  <<<(eth2 + B - 1) / B, B, 0, stream>>>(ei, ASRC2, ADST2, M2, S2, 1, eth2);
    edge_aggregate<<<(eth2 + B - 1) / B, B, 0, stream>>>(ei, ASRC2, ADST2, M2, S2,
                                                         H2, AGG2, 1, 128, eth2);

    // ===== global mean pool =====
    fill_f32<<<(G * 128 + B - 1) / B, B, 0, stream>>>(POOL, 0.f, G * 128);
    fill_f32<<<(G + B - 1) / B, B, 0, stream>>>(PCNT, 0.f, G);
    pool_accum<<<(N * 128 + B - 1) / B, B, 0, stream>>>(AGG2, b2, batch, POOL, N, 128);
    count_nodes<<<(N + B - 1) / B, B, 0, stream>>>(batch, PCNT, N);
    pool_div<<<(G * 128 + B - 1) / B, B, 0, stream>>>(POOL, PCNT, (float*)d_out,
                                                      G * 128, 128);
}